// DIRTBlock_28930899706574
// MI455X (gfx1250) — compile-verified
//
#include <hip/hip_runtime.h>
#include <math.h>

// ---------------- problem constants ----------------
#define DIMC 192      // channels
#define HEADS 8
#define HD 24         // DIMC / HEADS
#define TIMED 256     // time-embedding dim
#define HIDN 510      // GDFN hidden
#define BATCH 4
#define HH 128
#define WW 128
#define LL 16384      // HH*WW

typedef float v2f __attribute__((ext_vector_type(2)));
typedef float v8f __attribute__((ext_vector_type(8)));

// Exact fp32 WMMA: D(16x16) = A(16x4) * B(4x16) + C   [V_WMMA_F32_16X16X4_F32]
__device__ __forceinline__ v8f wmma4(v2f a, v2f b, v8f c) {
  return __builtin_amdgcn_wmma_f32_16x16x4_f32(
      /*neg_a=*/false, a, /*neg_b=*/false, b,
      /*c_mod=*/(short)0, c, /*reuse_a=*/false, /*reuse_b=*/false);
}

#define MT 4   // M tiles per wave  -> 64 output rows
#define NT 4   // N tiles per wave  -> 64 output cols

// ---------------------------------------------------------------------------
// GEMM: out[O x LL] = W[O x K] @ X[K x LL] + bias (+ optional residual)
// block = 128 threads = 4 waves; wave tile = 64(M) x 64(N); block = 64M x 256N
// grid = (LL/256, ceil(O/64))
// 16 WMMA per 16 loaded floats/lane per k-step; hot loop fully branchless.
// ---------------------------------------------------------------------------
__device__ __forceinline__ void load_frags(const float* const wr[MT],
                                           const float* __restrict__ Xc,
                                           int k0, v2f a[MT], v2f b[NT])
{
#pragma unroll
  for (int mt = 0; mt < MT; ++mt) {
    a[mt].x = wr[mt][k0];
    a[mt].y = wr[mt][k0 + 1];
  }
  const float* xp = Xc + (size_t)k0 * LL;
#pragma unroll
  for (int nt = 0; nt < NT; ++nt) {
    b[nt].x = xp[nt * 16];
    b[nt].y = xp[nt * 16 + LL];
  }
}

__global__ __launch_bounds__(128) void gemm_wmma_kernel(
    const float* __restrict__ W, const float* __restrict__ X,
    const float* __restrict__ bias, const float* __restrict__ res,
    float* __restrict__ out, int O, int K)
{
  const int lane  = threadIdx.x & 31;
  const int wv    = threadIdx.x >> 5;
  const int lm    = lane & 15;
  const int hi    = lane >> 4;          // 0: K offsets {0,1}, 1: {2,3}
  const int mBase = blockIdx.y * (MT * 16);
  const int nBase = blockIdx.x * 256 + wv * 64;

  // A-row pointers, clamped once (invalid rows pollute only unstored outputs)
  const float* wr[MT];
#pragma unroll
  for (int mt = 0; mt < MT; ++mt) {
    const int row = mBase + mt * 16 + lm;
    wr[mt] = W + (size_t)(row < O ? row : O - 1) * K;
  }
  const float* Xc = X + nBase + lm;

  v8f acc[MT][NT] = {};                  // 16 accumulators = 128 VGPRs

  const int Kmain = K & ~3;
  const int kOff  = 2 * hi;

  v2f a[MT], b[NT];
  load_frags(wr, Xc, kOff, a, b);

  for (int k = 0; k + 4 < Kmain; k += 4) {
    v2f na[MT], nb[NT];
    load_frags(wr, Xc, k + 4 + kOff, na, nb);   // prefetch next step
#pragma unroll
    for (int mt = 0; mt < MT; ++mt)
#pragma unroll
      for (int nt = 0; nt < NT; ++nt)
        acc[mt][nt] = wmma4(a[mt], b[nt], acc[mt][nt]);
#pragma unroll
    for (int mt = 0; mt < MT; ++mt) a[mt] = na[mt];
#pragma unroll
    for (int nt = 0; nt < NT; ++nt) b[nt] = nb[nt];
  }
#pragma unroll
  for (int mt = 0; mt < MT; ++mt)
#pragma unroll
    for (int nt = 0; nt < NT; ++nt)
      acc[mt][nt] = wmma4(a[mt], b[nt], acc[mt][nt]);

  if (Kmain < K) {                       // masked tail (only K=510): k = Kmain
    const int k0 = Kmain + kOff;
    const int k1 = k0 + 1;
    const float m0 = (k0 < K) ? 1.0f : 0.0f;   // lane-divergent cndmask
    const float m1 = (k1 < K) ? 1.0f : 0.0f;
    const int c0 = (k0 < K) ? k0 : K - 1;      // clamped, loads unconditional
    const int c1 = (k1 < K) ? k1 : K - 1;
    v2f ta[MT], tb[NT];
#pragma unroll
    for (int mt = 0; mt < MT; ++mt) {
      ta[mt].x = wr[mt][c0] * m0;
      ta[mt].y = wr[mt][c1] * m1;
    }
    const float* xp0 = Xc + (size_t)c0 * LL;
    const float* xp1 = Xc + (size_t)c1 * LL;
#pragma unroll
    for (int nt = 0; nt < NT; ++nt) {
      tb[nt].x = xp0[nt * 16] * m0;
      tb[nt].y = xp1[nt * 16] * m1;
    }
#pragma unroll
    for (int mt = 0; mt < MT; ++mt)
#pragma unroll
      for (int nt = 0; nt < NT; ++nt)
        acc[mt][nt] = wmma4(ta[mt], tb[nt], acc[mt][nt]);
  }

  // epilogue: C/D layout -> VGPR r holds (M = rowBase + r), rowBase += 8*hi
  const bool fullM = (mBase + MT * 16 <= O);   // uniform
  if (fullM) {
    if (res) {
#pragma unroll
      for (int mt = 0; mt < MT; ++mt) {
        const int rowBase = mBase + mt * 16 + hi * 8;
#pragma unroll
        for (int nt = 0; nt < NT; ++nt) {
          const int col = nBase + nt * 16 + lm;
#pragma unroll
          for (int r = 0; r < 8; ++r) {
            const int o = rowBase + r;
            out[(size_t)o * LL + col] =
                acc[mt][nt][r] + bias[o] + res[(size_t)o * LL + col];
          }
        }
      }
    } else {
#pragma unroll
      for (int mt = 0; mt < MT; ++mt) {
        const int rowBase = mBase + mt * 16 + hi * 8;
#pragma unroll
        for (int nt = 0; nt < NT; ++nt) {
          const int col = nBase + nt * 16 + lm;
#pragma unroll
          for (int r = 0; r < 8; ++r) {
            const int o = rowBase + r;
            out[(size_t)o * LL + col] = acc[mt][nt][r] + bias[o];
          }
        }
      }
    }
  } else {
#pragma unroll
    for (int mt = 0; mt < MT; ++mt) {
      const int rowBase = mBase + mt * 16 + hi * 8;
#pragma unroll
      for (int nt = 0; nt < NT; ++nt) {
        const int col = nBase + nt * 16 + lm;
#pragma unroll
        for (int r = 0; r < 8; ++r) {
          const int o = rowBase + r;
          if (o < O) {
            float val = acc[mt][nt][r] + bias[o];
            if (res) val += res[(size_t)o * LL + col];
            out[(size_t)o * LL + col] = val;
          }
        }
      }
    }
  }
}

// ---------------------------------------------------------------------------
// time MLP: out[b][j] = silu(temb[b]) . w[j] + b[j]   (j < 384)
// ---------------------------------------------------------------------------
__global__ void time_mlp_kernel(const float* __restrict__ temb,
                                const float* __restrict__ w,
                                const float* __restrict__ b,
                                float* __restrict__ out)
{
  __shared__ float s[TIMED];
  const int bb = blockIdx.y;
  for (int i = threadIdx.x; i < TIMED; i += blockDim.x) {
    const float v = temb[bb * TIMED + i];
    s[i] = v / (1.0f + expf(-v));
  }
  __syncthreads();
  const int j = blockIdx.x * blockDim.x + threadIdx.x;   // < 2*DIMC
  float acc = b[j];
  for (int i = 0; i < TIMED; ++i) acc += s[i] * w[j * TIMED + i];
  out[bb * (2 * DIMC) + j] = acc;
}

// ---------------------------------------------------------------------------
// channel LayerNorm (unbiased std) + scale/shift modulation
// one thread per pixel; grid = LL/256, block = 256
// ---------------------------------------------------------------------------
__global__ void ln_mod_kernel(const float* __restrict__ x,
                              const float* __restrict__ lnw,
                              const float* __restrict__ lnb,
                              const float* __restrict__ t,   // [2*DIMC] for this batch
                              float* __restrict__ out)
{
  const int l = blockIdx.x * blockDim.x + threadIdx.x;
  float sum = 0.0f, sq = 0.0f;
  for (int c = 0; c < DIMC; ++c) {
    const float v = x[c * LL + l];
    sum += v; sq += v * v;
  }
  const float mean = sum * (1.0f / DIMC);
  float var = (sq - (float)DIMC * mean * mean) * (1.0f / (DIMC - 1));
  var = fmaxf(var, 0.0f);
  const float inv = 1.0f / (sqrtf(var) + 1e-6f);
  for (int c = 0; c < DIMC; ++c) {
    const float v  = x[c * LL + l];
    const float xn = lnw[c] * (v - mean) * inv + lnb[c];
    out[c * LL + l] = xn * (1.0f + t[c]) + t[DIMC + c];
  }
}

// ---------------------------------------------------------------------------
// depthwise 3x3 SAME conv + bias; branchless borders (clamp + 0/1 mask)
// grid = (1, HH, C), block = WW
// ---------------------------------------------------------------------------
__global__ void dwconv3_kernel(const float* __restrict__ in,
                               const float* __restrict__ w,
                               const float* __restrict__ bias,
                               float* __restrict__ out)
{
  const int c = blockIdx.z, y = blockIdx.y, x = threadIdx.x;
  const float* ic = in + (size_t)c * LL;
  const float* wc = w + c * 9;
  float acc = bias[c];
#pragma unroll
  for (int dy = -1; dy <= 1; ++dy) {
    const int yy = y + dy;
    const int yc = min(max(yy, 0), HH - 1);
    const float my = (yy >= 0 && yy < HH) ? 1.0f : 0.0f;
#pragma unroll
    for (int dx = -1; dx <= 1; ++dx) {
      const int xx = x + dx;
      const int xc = min(max(xx, 0), WW - 1);
      const float m = (xx >= 0 && xx < WW) ? my : 0.0f;
      acc += ic[yc * WW + xc] * wc[(dy + 1) * 3 + (dx + 1)] * m;
    }
  }
  out[(size_t)c * LL + y * WW + x] = acc;
}

// depthwise 3x3 on both GDFN halves + gate x1*x2; grid = (1, HH, HIDN)
__global__ void dwgate_kernel(const float* __restrict__ h,
                              const float* __restrict__ w,
                              const float* __restrict__ bias,
                              float* __restrict__ g)
{
  const int c = blockIdx.z, y = blockIdx.y, x = threadIdx.x;
  const float* h1 = h + (size_t)c * LL;
  const float* h2 = h + (size_t)(c + HIDN) * LL;
  const float* w1 = w + c * 9;
  const float* w2 = w + (c + HIDN) * 9;
  float a1 = bias[c], a2 = bias[c + HIDN];
#pragma unroll
  for (int dy = -1; dy <= 1; ++dy) {
    const int yy = y + dy;
    const int yc = min(max(yy, 0), HH - 1);
    const float my = (yy >= 0 && yy < HH) ? 1.0f : 0.0f;
#pragma unroll
    for (int dx = -1; dx <= 1; ++dx) {
      const int xx = x + dx;
      const int xc = min(max(xx, 0), WW - 1);
      const float m = (xx >= 0 && xx < WW) ? my : 0.0f;
      const int off = yc * WW + xc;
      const int wi  = (dy + 1) * 3 + (dx + 1);
      a1 += h1[off] * w1[wi] * m;
      a2 += h2[off] * w2[wi] * m;
    }
  }
  g[(size_t)c * LL + y * WW + x] = a1 * a2;
}

// ---------------------------------------------------------------------------
// per-(head): Gram G = Q K^T (24x24 over L=16384), L2 norms, softmax -> attn
// block = 640 threads; LDS chunked, stride 257 -> conflict-free
// ---------------------------------------------------------------------------
#define ACH 256
__global__ __launch_bounds__(640) void attn_gram_kernel(
    const float* __restrict__ qkv,     // per-batch base of qkvd
    const float* __restrict__ temp,    // [HEADS]
    float* __restrict__ attn)          // [HEADS, 24, 24]
{
  __shared__ float sh[48 * (ACH + 1)];
  __shared__ float Gs[HD * HD];
  __shared__ float ns[48];
  const int h = blockIdx.x;
  const int t = threadIdx.x;
  const float* qb = qkv + (size_t)(h * HD) * LL;           // q channels
  const float* kb = qkv + (size_t)(DIMC + h * HD) * LL;    // k channels

  int i = 0, j = 0, r = 0;
  if (t < 576) { i = t / HD; j = t % HD; } else { r = t - 576; }
  float acc = 0.0f;

  for (int l0 = 0; l0 < LL; l0 += ACH) {
    for (int idx = t; idx < 48 * ACH; idx += 640) {
      const int rr = idx >> 8, cc = idx & (ACH - 1);
      const float* src = (rr < HD) ? (qb + (size_t)rr * LL)
                                   : (kb + (size_t)(rr - HD) * LL);
      sh[rr * (ACH + 1) + cc] = src[l0 + cc];
    }
    __syncthreads();
    if (t < 576) {
      const float* qi = &sh[i * (ACH + 1)];
      const float* kj = &sh[(HD + j) * (ACH + 1)];
      for (int cc = 0; cc < ACH; ++cc) acc += qi[cc] * kj[cc];
    } else if (r < 48) {
      const float* rw = &sh[r * (ACH + 1)];
      for (int cc = 0; cc < ACH; ++cc) acc += rw[cc] * rw[cc];
    }
    __syncthreads();
  }

  if (t < 576) Gs[i * HD + j] = acc;
  else if (r < 48) ns[r] = acc;
  __syncthreads();

  if (t < HD) {
    const float scale = temp[h];
    const float nq = fmaxf(sqrtf(ns[t]), 1e-12f);
    float row[HD];
    float mx = -1e30f;
    for (int jj = 0; jj < HD; ++jj) {
      const float nk = fmaxf(sqrtf(ns[HD + jj]), 1e-12f);
      const float s = Gs[t * HD + jj] / (nq * nk) * scale;
      row[jj] = s;
      mx = fmaxf(mx, s);
    }
    float sum = 0.0f;
    for (int jj = 0; jj < HD; ++jj) { row[jj] = expf(row[jj] - mx); sum += row[jj]; }
    const float isum = 1.0f / sum;
    for (int jj = 0; jj < HD; ++jj)
      attn[(h * HD + t) * HD + jj] = row[jj] * isum;
  }
}

// out[h*24+c][l] = sum_d attn[h][c][d] * v[h*24+d][l]; grid = (LL/256, HEADS)
__global__ void attnv_kernel(const float* __restrict__ v,   // qkvd + 2*DIMC*LL
                             const float* __restrict__ attn,
                             float* __restrict__ out)
{
  __shared__ float as[HD * HD];
  const int h = blockIdx.y;
  for (int idx = threadIdx.x; idx < HD * HD; idx += blockDim.x)
    as[idx] = attn[h * HD * HD + idx];
  __syncthreads();
  const int l = blockIdx.x * blockDim.x + threadIdx.x;
  const float* vb = v + (size_t)(h * HD) * LL;
  float vv[HD];
#pragma unroll
  for (int d = 0; d < HD; ++d) vv[d] = vb[(size_t)d * LL + l];
#pragma unroll
  for (int c = 0; c < HD; ++c) {
    float acc = 0.0f;
#pragma unroll
    for (int d = 0; d < HD; ++d) acc += as[c * HD + d] * vv[d];
    out[(size_t)(h * HD + c) * LL + l] = acc;
  }
}

// ---------------------------------------------------------------------------
extern "C" void kernel_launch(void* const* d_in, const int* in_sizes, int n_in,
                              void* d_out, int out_size, void* d_ws, size_t ws_size,
                              hipStream_t stream)
{
  const float* x        = (const float*)d_in[0];
  const float* temb     = (const float*)d_in[1];
  const float* n1_w     = (const float*)d_in[2];
  const float* n1_b     = (const float*)d_in[3];
  const float* temp     = (const float*)d_in[4];
  const float* a_qkv_w  = (const float*)d_in[5];
  const float* a_qkv_b  = (const float*)d_in[6];
  const float* a_dw_w   = (const float*)d_in[7];
  const float* a_dw_b   = (const float*)d_in[8];
  const float* a_proj_w = (const float*)d_in[9];
  const float* a_proj_b = (const float*)d_in[10];
  const float* a_t_w    = (const float*)d_in[11];
  const float* a_t_b    = (const float*)d_in[12];
  const float* n2_w     = (const float*)d_in[13];
  const float* n2_b     = (const float*)d_in[14];
  const float* f_c1_w   = (const float*)d_in[15];
  const float* f_c1_b   = (const float*)d_in[16];
  const float* f_dw_w   = (const float*)d_in[17];
  const float* f_dw_b   = (const float*)d_in[18];
  const float* f_c2_w   = (const float*)d_in[19];
  const float* f_c2_b   = (const float*)d_in[20];
  const float* f_t_w    = (const float*)d_in[21];
  const float* f_t_b    = (const float*)d_in[22];
  float* out = (float*)d_out;

  // per-batch workspace layout (~113 MB total, slots reused across phases)
  char* ws = (char*)d_ws;
  constexpr size_t SZ_S0  = (size_t)DIMC * LL * 4;      // 12.6 MB (xm / ym)
  constexpr size_t SZ_QKV = (size_t)3 * DIMC * LL * 4;  // 37.7 MB
  constexpr size_t S0 = 0;
  constexpr size_t S1 = S0 + SZ_S0;                     // qkv   (later: h)
  constexpr size_t S2 = S1 + SZ_QKV;                    // qkvd
  constexpr size_t S3 = S2 + SZ_QKV;                    // attn-out
  constexpr size_t OFF_H = S1;                          // h: 1020*LL*4 = 66.8 MB
  constexpr size_t OFF_G = OFF_H + (size_t)2 * HIDN * LL * 4;  // g: 33.4 MB
  constexpr size_t OFF_T = OFF_G + (size_t)HIDN * LL * 4;      // tiny tail

  float* xm   = (float*)(ws + S0);
  float* qkv  = (float*)(ws + S1);
  float* qkvd = (float*)(ws + S2);
  float* aout = (float*)(ws + S3);
  float* hbuf = (float*)(ws + OFF_H);
  float* gbuf = (float*)(ws + OFF_G);
  float* ta   = (float*)(ws + OFF_T);
  float* tf   = ta + BATCH * 2 * DIMC;
  float* attn = tf + BATCH * 2 * DIMC;

  // time-conditioning MLPs (all batches, tiny)
  time_mlp_kernel<<<dim3(3, BATCH), 128, 0, stream>>>(temb, a_t_w, a_t_b, ta);
  time_mlp_kernel<<<dim3(3, BATCH), 128, 0, stream>>>(temb, f_t_w, f_t_b, tf);

  for (int b = 0; b < BATCH; ++b) {
    const float* xb = x   + (size_t)b * DIMC * LL;
    float*       ob = out + (size_t)b * DIMC * LL;

    // ---- MDTA ----
    ln_mod_kernel<<<LL / 256, 256, 0, stream>>>(xb, n1_w, n1_b, ta + b * 2 * DIMC, xm);
    gemm_wmma_kernel<<<dim3(LL / 256, 9), 128, 0, stream>>>(
        a_qkv_w, xm, a_qkv_b, nullptr, qkv, 3 * DIMC, DIMC);
    dwconv3_kernel<<<dim3(1, HH, 3 * DIMC), WW, 0, stream>>>(qkv, a_dw_w, a_dw_b, qkvd);
    attn_gram_kernel<<<HEADS, 640, 0, stream>>>(qkvd, temp, attn);
    attnv_kernel<<<dim3(LL / 256, HEADS), 256, 0, stream>>>(
        qkvd + (size_t)2 * DIMC * LL, attn, aout);
    gemm_wmma_kernel<<<dim3(LL / 256, 3), 128, 0, stream>>>(
        a_proj_w, aout, a_proj_b, /*res=*/xb, ob, DIMC, DIMC);

    // ---- GDFN ----
    ln_mod_kernel<<<LL / 256, 256, 0, stream>>>(ob, n2_w, n2_b, tf + b * 2 * DIMC, xm);
    gemm_wmma_kernel<<<dim3(LL / 256, 16), 128, 0, stream>>>(
        f_c1_w, xm, f_c1_b, nullptr, hbuf, 2 * HIDN, DIMC);
    dwgate_kernel<<<dim3(1, HH, HIDN), WW, 0, stream>>>(hbuf, f_dw_w, f_dw_b, gbuf);
    gemm_wmma_kernel<<<dim3(LL / 256, 3), 128, 0, stream>>>(
        f_c2_w, gbuf, f_c2_b, /*res=*/ob, ob, DIMC, HIDN);
  }
}